// snn_1_47725676593665
// MI455X (gfx1250) — compile-verified
//
#include <hip/hip_runtime.h>
#include <hip/hip_bf16.h>

typedef __attribute__((ext_vector_type(2))) float v2f;
typedef __attribute__((ext_vector_type(8))) float v8f;

// Problem constants
#define TB_M      4992            // T*B*39
#define KIN       2209            // 47*47
#define KIN_PAD   2240            // padded to 70*32
#define H0        2048
#define H1        1024
#define MROWS     624             // B*39 = 16*39
#define MROWS_PAD 640             // padded to 10*64 for unpredicated A loads
#define OUTFEAT   39936           // 39*1024
#define LDST      44              // LDS row stride (floats): 16B-aligned rows, bank-conflict-free

// ---------------------------------------------------------------- utilities
__global__ void zero_kernel(float* __restrict__ p, int n) {
    int i = blockIdx.x * 256 + threadIdx.x;
    if (i < n) p[i] = 0.0f;
}

// ---------------------------------------------------------------- pooling
// xs[m][f], m = ((t*16+b)*39+n), f = hp*47+wp ; row padded to 2240 with zeros
__global__ void pool_kernel(const float* __restrict__ x, float* __restrict__ xs) {
    long idx = (long)blockIdx.x * 256 + threadIdx.x;
    if (idx >= (long)TB_M * KIN_PAD) return;
    int m = (int)(idx / KIN_PAD);
    int f = (int)(idx % KIN_PAD);
    if (f >= KIN) { xs[idx] = 0.0f; return; }
    int t  = m / 624;
    int r  = m % 624;
    int b  = r / 39;
    int n  = r % 39;
    int c  = n / 13;
    int dp = n % 13;
    int hp = f / 47, wp = f % 47;
    int d0 = t * 29 + dp * 2;
    const float* base = x + ((((long)b * 3 + c) * 232 + d0) * 96 + hp * 2) * 96 + wp * 2;
    float s = 0.0f;
#pragma unroll
    for (int kd = 0; kd < 4; ++kd) {
#pragma unroll
        for (int kh = 0; kh < 4; ++kh) {
            const float* p = base + (long)kd * 9216 + kh * 96;
            s += p[0] + p[1] + p[2] + p[3];
        }
    }
    xs[idx] = s * (1.0f / 64.0f);
}

// ---------------------------------------------------------------- weight pack
// Wp[h][2240] = W_in[h][0..2208], zero tail -> all GEMM1 B loads unpredicated b128
__global__ void pack_kernel(const float* __restrict__ W, float* __restrict__ Wp) {
    long idx = (long)blockIdx.x * 256 + threadIdx.x;
    if (idx >= (long)H0 * KIN_PAD) return;
    int h = (int)(idx / KIN_PAD);
    int f = (int)(idx % KIN_PAD);
    Wp[idx] = (f < KIN) ? W[(long)h * KIN + f] : 0.0f;
}

// ---------------------------------------------------------------- WMMA GEMM
// C[m][n] = sum_k A[m][k] * B[n][k] + bias[n]
// Tile 64x64, BK=32, 128 threads (4 wave32s), fp32 WMMA 16x16x4 (exact fp32).
// Double-buffered LDS; tiles staged with GLOBAL_LOAD_ASYNC_TO_LDS_B128 and
// synchronized with s_wait_asynccnt (in-order async completion) + s_barrier.
// A and B rows must be readable (zero/garbage-padded) over k in [0, ktiles*32)
// and over the full 64-row tiles (callers pad M / pack K).
__global__ __launch_bounds__(128) void gemm_bias_kernel(
    const float* __restrict__ A, int lda,
    const float* __restrict__ B, int ldb,
    const float* __restrict__ bias,
    float* __restrict__ C, int ldc,
    int M, int ktiles) {

    __shared__ float As[2][64 * LDST];
    __shared__ float Bs[2][64 * LDST];

    const int tid  = threadIdx.x;
    const int lane = tid & 31;
    const int wave = tid >> 5;
    const int m0 = blockIdx.y * 64;
    const int n0 = blockIdx.x * 64;

    // WMMA f32 16x16x4 fragment addressing (ISA 7.12.2, 32-bit A 16x4):
    // lane l holds (row = l&15, k = 2*(l>>4) + {0,1})
    const int row_l = lane & 15;
    const int ksel  = (lane >> 4) * 2;

    // staging map: 128 threads -> 16 rows x 8 k-quads; 4 row-passes, b128 each
    const int lr = tid >> 3;        // 0..15
    const int lk = (tid & 7) * 4;   // 0,4,...,28

    const float* gA = A + (long)(m0 + lr) * lda + lk;
    const float* gB = B + (long)(n0 + lr) * ldb + lk;
    const unsigned laA0 = (unsigned)(unsigned long)&As[0][lr * LDST + lk];
    const unsigned laB0 = (unsigned)(unsigned long)&Bs[0][lr * LDST + lk];
    const unsigned bufStride = (unsigned)(64 * LDST * 4); // bytes between buffers

    // issue one k-tile's async loads (8 x b128 per thread)
    auto issue_tile = [&](int kt, int buf) {
        const float* a = gA + kt * 32;
        const float* b = gB + kt * 32;
        unsigned la = laA0 + (unsigned)buf * bufStride;
        unsigned lb = laB0 + (unsigned)buf * bufStride;
#pragma unroll
        for (int rr = 0; rr < 4; ++rr) {
            asm volatile("global_load_async_to_lds_b128 %0, %1, off"
                         :: "v"(la + (unsigned)(rr * 16 * LDST * 4)),
                            "v"((unsigned long)(a + (long)rr * 16 * lda))
                         : "memory");
            asm volatile("global_load_async_to_lds_b128 %0, %1, off"
                         :: "v"(lb + (unsigned)(rr * 16 * LDST * 4)),
                            "v"((unsigned long)(b + (long)rr * 16 * ldb))
                         : "memory");
        }
    };

    v8f acc[4];
#pragma unroll
    for (int j = 0; j < 4; ++j) acc[j] = (v8f){0, 0, 0, 0, 0, 0, 0, 0};

    issue_tile(0, 0);

    for (int kt = 0; kt < ktiles; ++kt) {
        const int buf = kt & 1;
        if (kt + 1 < ktiles) {
            issue_tile(kt + 1, buf ^ 1);
            // oldest 8 (tile kt) done once <=8 remain outstanding (in-order)
            asm volatile("s_wait_asynccnt 8" ::: "memory");
        } else {
            asm volatile("s_wait_asynccnt 0" ::: "memory");
        }
        __syncthreads();

        const float* arow = &As[buf][(wave * 16 + row_l) * LDST + ksel];
        const float* brow = &Bs[buf][row_l * LDST + ksel];
#pragma unroll
        for (int kk = 0; kk < 32; kk += 4) {
            v2f a = *(const v2f*)(arow + kk);
#pragma unroll
            for (int j = 0; j < 4; ++j) {
                v2f b = *(const v2f*)(brow + j * 16 * LDST + kk);
                acc[j] = __builtin_amdgcn_wmma_f32_16x16x4_f32(
                    false, a, false, b, (short)0, acc[j], false, false);
            }
        }
        __syncthreads();   // readers done before next issue overwrites this buf
    }

    // epilogue: C/D layout — VGPR r of lane l = C[r + 8*(l>>4)][l&15]
    const int hi   = lane >> 4;
    const int colb = lane & 15;
#pragma unroll
    for (int j = 0; j < 4; ++j) {
        int gn = n0 + j * 16 + colb;
        float bv = bias[gn];
#pragma unroll
        for (int r = 0; r < 8; ++r) {
            int gm = m0 + wave * 16 + hi * 8 + r;
            if (gm < M) C[(long)gm * ldc + gn] = acc[j][r] + bv;
        }
    }
}

// ---------------------------------------------------------------- LIF step
// snntorch Leaky, reset_mechanism='subtract', thr=1:
//   reset = (mem - 1 > 0); mem' = 0.9*mem + cur - reset; spk = (mem' - 1 > 0)
__global__ void lif_kernel(const float* __restrict__ cur,
                           float* __restrict__ mem,
                           float* __restrict__ spk, int n) {
    int i = blockIdx.x * 256 + threadIdx.x;
    if (i >= n) return;
    float m = mem[i];
    float reset = (m - 1.0f > 0.0f) ? 1.0f : 0.0f;
    float mn = 0.9f * m + cur[i] - reset;
    mem[i] = mn;
    spk[i] = (mn - 1.0f > 0.0f) ? 1.0f : 0.0f;
}

// ---------------------------------------------------------------- fc_out + output LIF
// one block per (b, class j); dot(spk_hid[b,:], W_out[j,:]) + b_out, then LIF
__global__ __launch_bounds__(256) void fcout_lif_kernel(
    const float* __restrict__ spk_hid, const float* __restrict__ W_out,
    const float* __restrict__ b_out, float* __restrict__ m_out,
    float* __restrict__ out_t) {
    __shared__ float red[256];
    int blk = blockIdx.x;           // b*4 + j
    int b = blk >> 2, j = blk & 3;
    const float* sp = spk_hid + (long)b * OUTFEAT;
    const float* w  = W_out   + (long)j * OUTFEAT;
    float s = 0.0f;
    for (int i = threadIdx.x; i < OUTFEAT; i += 256) s += sp[i] * w[i];
    red[threadIdx.x] = s;
    __syncthreads();
    for (int off = 128; off > 0; off >>= 1) {
        if (threadIdx.x < off) red[threadIdx.x] += red[threadIdx.x + off];
        __syncthreads();
    }
    if (threadIdx.x == 0) {
        float cur = red[0] + b_out[j];
        float m = m_out[blk];
        float reset = (m - 1.0f > 0.0f) ? 1.0f : 0.0f;
        float mn = 0.9f * m + cur - reset;
        m_out[blk] = mn;
        out_t[blk] = (mn - 1.0f > 0.0f) ? 1.0f : 0.0f;
    }
}

// ---------------------------------------------------------------- launch
extern "C" void kernel_launch(void* const* d_in, const int* in_sizes, int n_in,
                              void* d_out, int out_size, void* d_ws, size_t ws_size,
                              hipStream_t stream) {
    const float* x     = (const float*)d_in[0];
    const float* W_in  = (const float*)d_in[1];
    const float* b_in  = (const float*)d_in[2];
    const float* W_hid = (const float*)d_in[3];
    const float* b_hid = (const float*)d_in[4];
    const float* W_out = (const float*)d_in[5];
    const float* b_out = (const float*)d_in[6];
    float* out = (float*)d_out;

    float* ws = (float*)d_ws;
    size_t off = 0;
    float* xs      = ws + off; off += (size_t)TB_M * KIN_PAD;       // 11.18M
    float* Wp      = ws + off; off += (size_t)H0 * KIN_PAD;         //  4.59M
    float* cur_all = ws + off; off += (size_t)TB_M * H0;            // 10.22M
    float* m_in    = ws + off; off += (size_t)MROWS * H0;
    float* spk_in  = ws + off; off += (size_t)MROWS_PAD * H0;       // padded rows
    float* cur_hid = ws + off; off += (size_t)MROWS * H1;
    float* m_hid   = ws + off; off += (size_t)MROWS * H1;
    float* spk_hid = ws + off; off += (size_t)MROWS * H1;
    float* m_out   = ws + off; off += 64;

    // zero LIF states + spk_in pad rows every call (graph replays re-run this)
    {
        int n1 = MROWS * H0;
        int n2 = MROWS * H1;
        int np = (MROWS_PAD - MROWS) * H0;
        zero_kernel<<<(n1 + 255) / 256, 256, 0, stream>>>(m_in, n1);
        zero_kernel<<<(n2 + 255) / 256, 256, 0, stream>>>(m_hid, n2);
        zero_kernel<<<(np + 255) / 256, 256, 0, stream>>>(spk_in + (size_t)MROWS * H0, np);
        zero_kernel<<<1, 64, 0, stream>>>(m_out, 64);
    }

    // pooling -> xs (zero-padded K); pack W_in -> Wp (zero-padded K)
    {
        long total = (long)TB_M * KIN_PAD;
        pool_kernel<<<(unsigned)((total + 255) / 256), 256, 0, stream>>>(x, xs);
        long wtotal = (long)H0 * KIN_PAD;
        pack_kernel<<<(unsigned)((wtotal + 255) / 256), 256, 0, stream>>>(W_in, Wp);
    }

    // fc_in for all timesteps: cur_all[m][h], m t-major
    {
        dim3 grid(H0 / 64, TB_M / 64);
        gemm_bias_kernel<<<grid, 128, 0, stream>>>(
            xs, KIN_PAD, Wp, KIN_PAD, b_in, cur_all, H0,
            TB_M, KIN_PAD / 32);
    }

    // sequential timesteps
    for (int t = 0; t < 8; ++t) {
        const float* cur_t = cur_all + (size_t)t * MROWS * H0;
        int nin = MROWS * H0;
        lif_kernel<<<(nin + 255) / 256, 256, 0, stream>>>(cur_t, m_in, spk_in, nin);

        dim3 grid2(H1 / 64, MROWS_PAD / 64);
        gemm_bias_kernel<<<grid2, 128, 0, stream>>>(
            spk_in, H0, W_hid, H0, b_hid, cur_hid, H1,
            MROWS, H0 / 32);

        int nhid = MROWS * H1;
        lif_kernel<<<(nhid + 255) / 256, 256, 0, stream>>>(cur_hid, m_hid, spk_hid, nhid);

        fcout_lif_kernel<<<64, 256, 0, stream>>>(spk_hid, W_out, b_out, m_out, out + t * 64);
    }
}